// conv3x3_16484084483745
// MI455X (gfx1250) — compile-verified
//
#include <hip/hip_runtime.h>
#include <math.h>

typedef __attribute__((ext_vector_type(2))) float v2f;
typedef __attribute__((ext_vector_type(8))) float v8f;

#define Bn 8
#define Cc 256
#define Oc 256
#define Hh 64
#define Ww 64
#define K2 9
#define KK 2304      // C * K2 (GEMM K dimension)
#define KC 72        // K-chunk: 8 channels x 9 taps
#define KCP 74       // padded LDS stride (even -> 8B aligned b64, bank-spread)
#define NT 16        // pixel tile (GEMM N tile)
#define NCHUNK 32    // 2304 / 72

// ---------------------------------------------------------------------------
// Kernel 1: offset conv (18 output channels, padded to 32 = 2 WMMA o-tiles)
// out_off[b, j, h, w] = b_off[j] + sum_{c,tap} relu(x[b,c,h+dy,w+dx]) * w_off[j,c,tap]
// ---------------------------------------------------------------------------
__global__ __launch_bounds__(64) void offs_conv_wmma(
    const float* __restrict__ x, const float* __restrict__ w_off,
    const float* __restrict__ b_off, float* __restrict__ offs)
{
    __shared__ float bsm[NT * KCP];

    const int tile = blockIdx.x;          // 2048 tiles
    const int b    = tile >> 8;
    const int rem  = tile & 255;
    const int h    = rem >> 2;
    const int w0   = (rem & 3) << 4;
    const int tid  = threadIdx.x;         // 0..63
    const int lane = tid & 31;
    const int wv   = tid >> 5;            // 0..1 (o-tile per wave)
    const int ln16 = lane & 15;
    const int khalf= lane >> 4;           // 0/1 -> K offset 0/2

    const float* xb = x + b * (Cc * Hh * Ww);

    v8f acc = {};
    const int orow  = wv * 16 + ln16;     // 0..31 (rows >= 18 are padding)
    const bool arow_ok = (orow < 18);

    for (int ch = 0; ch < NCHUNK; ++ch) {
        const int kb = ch * KC;
        const int cb = ch * 8;
        __syncthreads();
        // cooperative im2col fill: 16 pixels x 72 k-values
        for (int e = tid; e < NT * KC; e += 64) {
            int kidx = e >> 4;
            int n    = e & 15;
            int c    = cb + kidx / 9;
            int tap  = kidx % 9;
            int y    = h + (tap / 3) - 1;
            int xx   = w0 + n + (tap % 3) - 1;
            float v  = 0.f;
            if (y >= 0 && y < Hh && xx >= 0 && xx < Ww) {
                float t = xb[c * (Hh * Ww) + y * Ww + xx];
                v = t > 0.f ? t : 0.f;                 // fused relu
            }
            bsm[n * KCP + kidx] = v;
        }
        __syncthreads();
        #pragma unroll
        for (int s = 0; s < KC / 4; ++s) {
            int kc = kb + 4 * s + 2 * khalf;
            v2f a = {};
            if (arow_ok)
                a = *(const v2f*)&w_off[orow * KK + kc];   // B64, 8B aligned
            v2f bm = *(const v2f*)&bsm[ln16 * KCP + 4 * s + 2 * khalf];
            acc = __builtin_amdgcn_wmma_f32_16x16x4_f32(
                false, a, false, bm, (short)0, acc, false, false);
        }
    }
    // store valid rows (< 18) with bias
    #pragma unroll
    for (int v = 0; v < 8; ++v) {
        int m = v + 8 * khalf;
        int o = wv * 16 + m;
        if (o < 18)
            offs[(b * 18 + o) * (Hh * Ww) + h * Ww + w0 + ln16] = acc[v] + b_off[o];
    }
}

// ---------------------------------------------------------------------------
// Kernel 2: deformable conv as implicit-im2col GEMM with bilinear gather.
// One WG per 16-pixel tile; 8 waves cover all 256 output channels (2 tiles ea).
// ---------------------------------------------------------------------------
__global__ __launch_bounds__(256) void deform_gemm_wmma(
    const float* __restrict__ x, const float* __restrict__ offs,
    const float* __restrict__ w_def, float* __restrict__ out)
{
    __shared__ float bsm[NT * KCP];
    __shared__ int   s_y0[NT * K2];
    __shared__ int   s_x0[NT * K2];
    __shared__ float s_wy[NT * K2];
    __shared__ float s_wx[NT * K2];

    const int tile = blockIdx.x;          // 2048 tiles
    const int b    = tile >> 8;
    const int rem  = tile & 255;
    const int h    = rem >> 2;
    const int w0   = (rem & 3) << 4;
    const int tid  = threadIdx.x;         // 0..255
    const int lane = tid & 31;
    const int wv   = tid >> 5;            // 0..7
    const int ln16 = lane & 15;
    const int khalf= lane >> 4;

    const float* xb = x + b * (Cc * Hh * Ww);

    // Phase A: per-(pixel, tap) sample coords, shared across all 256 channels
    if (tid < NT * K2) {
        int n = tid / 9, tap = tid % 9;
        float dy = offs[(b * 18 + 2 * tap    ) * (Hh * Ww) + h * Ww + w0 + n];
        float dx = offs[(b * 18 + 2 * tap + 1) * (Hh * Ww) + h * Ww + w0 + n];
        float py = (float)(h - 1 + tap / 3) + dy;
        float px = (float)(w0 + n - 1 + tap % 3) + dx;
        float fy = floorf(py), fx = floorf(px);
        s_y0[tid] = (int)fy;  s_x0[tid] = (int)fx;
        s_wy[tid] = py - fy;  s_wx[tid] = px - fx;
    }
    // (covered by the first __syncthreads in the loop)

    const int oA   = wv * 16;
    const int rowA = oA + ln16;
    const int rowB = rowA + 128;

    v8f accA = {}, accB = {};

    for (int ch = 0; ch < NCHUNK; ++ch) {
        const int kb = ch * KC;
        const int cb = ch * 8;
        __syncthreads();
        // Phase B1: cooperative bilinear im2col fill (16 x 72), zero-pad OOB
        for (int e = tid; e < NT * KC; e += 256) {
            int kidx = e >> 4;
            int n    = e & 15;
            int c    = cb + kidx / 9;
            int tap  = kidx % 9;
            int p    = n * 9 + tap;
            int y0   = s_y0[p], x0 = s_x0[p];
            float wy = s_wy[p], wx = s_wx[p];
            const float* xc = xb + c * (Hh * Ww);
            float v00 = 0.f, v01 = 0.f, v10 = 0.f, v11 = 0.f;
            bool xl = (x0     >= 0) && (x0     < Ww);
            bool xr = (x0 + 1 >= 0) && (x0 + 1 < Ww);
            if (y0 >= 0 && y0 < Hh) {
                const float* r = xc + y0 * Ww;
                if (xl) { float t = r[x0];     v00 = t > 0.f ? t : 0.f; }
                if (xr) { float t = r[x0 + 1]; v01 = t > 0.f ? t : 0.f; }
            }
            if (y0 + 1 >= 0 && y0 + 1 < Hh) {
                const float* r = xc + (y0 + 1) * Ww;
                if (xl) { float t = r[x0];     v10 = t > 0.f ? t : 0.f; }
                if (xr) { float t = r[x0 + 1]; v11 = t > 0.f ? t : 0.f; }
            }
            float v = (v00 * (1.f - wy) + v10 * wy) * (1.f - wx)
                    + (v01 * (1.f - wy) + v11 * wy) * wx;
            bsm[n * KCP + kidx] = v;
        }
        __syncthreads();
        if (kb + KC < KK)
            __builtin_prefetch(&w_def[rowA * KK + kb + KC], 0, 1);
        // Phase B2: 18 WMMA K-steps x 2 o-tiles per wave
        #pragma unroll
        for (int s = 0; s < KC / 4; ++s) {
            int kc  = kb + 4 * s + 2 * khalf;
            v2f aA  = *(const v2f*)&w_def[rowA * KK + kc];   // global B64, L2-hot
            v2f aB  = *(const v2f*)&w_def[rowB * KK + kc];
            v2f bm  = *(const v2f*)&bsm[ln16 * KCP + 4 * s + 2 * khalf]; // ds b64
            accA = __builtin_amdgcn_wmma_f32_16x16x4_f32(
                false, aA, false, bm, (short)0, accA, false, false);
            accB = __builtin_amdgcn_wmma_f32_16x16x4_f32(
                false, aB, false, bm, (short)0, accB, false, false);
        }
    }

    // Phase C: store D tiles (M = v + 8*khalf, N = ln16)
    float* ob = out + b * (Oc * Hh * Ww) + h * Ww + w0 + ln16;
    #pragma unroll
    for (int v = 0; v < 8; ++v) {
        int m = v + 8 * khalf;
        ob[(oA + m)       * (Hh * Ww)] = accA[v];
        ob[(oA + m + 128) * (Hh * Ww)] = accB[v];
    }
}

// ---------------------------------------------------------------------------
extern "C" void kernel_launch(void* const* d_in, const int* in_sizes, int n_in,
                              void* d_out, int out_size, void* d_ws, size_t ws_size,
                              hipStream_t stream) {
    const float* x     = (const float*)d_in[0];  // [8,256,64,64]
    const float* w_off = (const float*)d_in[1];  // [18,256,3,3]
    const float* b_off = (const float*)d_in[2];  // [18]
    const float* w_def = (const float*)d_in[3];  // [256,256,3,3]
    float* out  = (float*)d_out;                 // [8,256,64,64]
    float* offw = (float*)d_ws;                  // 8*18*4096 floats = 2.36 MB

    offs_conv_wmma<<<dim3(2048), dim3(64), 0, stream>>>(x, w_off, b_off, offw);
    deform_gemm_wmma<<<dim3(2048), dim3(256), 0, stream>>>(x, offw, w_def, out);
}